// FALayer_4037269258322
// MI455X (gfx1250) — compile-verified
//
#include <hip/hip_runtime.h>
#include <hip/hip_bf16.h>

#define DFEAT 64

typedef float v2f __attribute__((ext_vector_type(2)));
typedef float v8f __attribute__((ext_vector_type(8)));

__device__ __forceinline__ float fast_tanh(float x) {
#if __has_builtin(__builtin_amdgcn_tanhf)
    return __builtin_amdgcn_tanhf(x);   // gfx1250 v_tanh_f32
#else
    return tanhf(x);
#endif
}

// ---------------------------------------------------------------------------
// Kernel 1: zero the output accumulator (harness poisons d_out with 0xAA).
// ---------------------------------------------------------------------------
__global__ void fa_zero_kernel(float* __restrict__ z, int total) {
    int i = blockIdx.x * blockDim.x + threadIdx.x;
    if (i < total) z[i] = 0.0f;
}

// ---------------------------------------------------------------------------
// Kernel 2: per-node gate projections via V_WMMA_F32_16X16X4_F32.
//   a[n] = dot(h[n], gate_w[0:64])     (w_dst)
//   b[n] = dot(h[n], gate_w[64:128])   (w_src)
// One wave handles 16 node rows; K=64 accumulated over 16 WMMA steps.
// A-matrix 16x4 layout (ISA 7.12.2): lane L -> M = L%16, half = L/16,
//   A.x = K=2*half, A.y = K=2*half+1 (within the 4-wide K chunk).
// B-matrix 4x16 layout: VGPR v -> K = v + 2*half, N = L%16.
// C/D 16x16 f32 layout: VGPR r -> M = r + 8*half, N = L%16.
// B is built branchlessly: per-lane base pointer (w_dst for N==0, w_src for
// N==1) chosen once, unconditional float2 load each step, scaled by a
// loop-invariant validity mask (0 for N>=2).
// ---------------------------------------------------------------------------
__global__ void fa_node_gates_kernel(const float* __restrict__ h,
                                     const float* __restrict__ gate_w,
                                     float* __restrict__ a,
                                     float* __restrict__ b,
                                     int n) {
    int wave = (blockIdx.x * blockDim.x + threadIdx.x) >> 5;
    int lane = threadIdx.x & 31;
    int base = wave * 16;              // uniform per wave
    if (base >= n) return;             // whole-wave exit: keeps EXEC all-1 for WMMA

    int m     = lane & 15;             // N column in B/C; M row (mod 16) in A
    int khalf = lane >> 4;             // 0 or 1

    const float2* hrow2 = (const float2*)(h + (size_t)(base + m) * DFEAT);

    // Loop-invariant B selection (no branches inside the WMMA loop).
    const float2* w2   = (const float2*)(gate_w + ((m == 1) ? DFEAT : 0));
    float         valid = (m < 2) ? 1.0f : 0.0f;

    v8f c = {};
    #pragma unroll
    for (int t = 0; t < 16; ++t) {
        int kq = 2 * t + khalf;        // float2 index: covers K = {2*kq, 2*kq+1}

        float2 av = hrow2[kq];
        v2f A; A.x = av.x; A.y = av.y;

        float2 wv = w2[kq];
        v2f B; B.x = wv.x * valid; B.y = wv.y * valid;

        c = __builtin_amdgcn_wmma_f32_16x16x4_f32(
                /*neg_a=*/false, A, /*neg_b=*/false, B,
                /*c_mod=*/(short)0, c, /*reuse_a=*/false, /*reuse_b=*/false);
    }

    // Extract column 0 -> a, column 1 -> b.
    int rbase = base + 8 * khalf;
    if (m == 0) {
        #pragma unroll
        for (int r = 0; r < 8; ++r) a[rbase + r] = c[r];
    } else if (m == 1) {
        #pragma unroll
        for (int r = 0; r < 8; ++r) b[rbase + r] = c[r];
    }
}

// ---------------------------------------------------------------------------
// Kernel 3: one wave32 per edge.
//   g  = tanh(a[dst] + b[src] + gate_b)
//   ev = g * dnorm[dst] * dnorm[src]
//   z[dst, :] += ev * h[src, :]        (64 f32 no-return L2 atomics per edge)
// Lane l covers feature dims {2l, 2l+1} via a float2 gather.
// Edge indices are wave-uniform -> readfirstlane them into SGPRs so the
// h[src] gather uses the saddr global-load form and gate math scalarizes.
// ---------------------------------------------------------------------------
__global__ void fa_edge_kernel(const float* __restrict__ h,
                               const float* __restrict__ dnorm,
                               const float* __restrict__ a,
                               const float* __restrict__ b,
                               const int* __restrict__ src,
                               const int* __restrict__ dst,
                               const float* __restrict__ gate_b,
                               float* __restrict__ z,
                               int e) {
    int wave = blockIdx.x * (blockDim.x >> 5) + (threadIdx.x >> 5);
    int lane = threadIdx.x & 31;
    if (wave >= e) return;

    int s = __builtin_amdgcn_readfirstlane(src[wave]);
    int d = __builtin_amdgcn_readfirstlane(dst[wave]);

    float g  = fast_tanh(a[d] + b[s] + gate_b[0]);
    float ev = g * dnorm[d] * dnorm[s];

    const float2* hs = (const float2*)(h + (size_t)s * DFEAT);
    float2 v = hs[lane];

    float* zd = z + (size_t)d * DFEAT + 2 * lane;
    atomicAdd(zd,     v.x * ev);
    atomicAdd(zd + 1, v.y * ev);
}

// ---------------------------------------------------------------------------
// Launch: zero z -> node gate projections (WMMA) -> edge scatter.
// ---------------------------------------------------------------------------
extern "C" void kernel_launch(void* const* d_in, const int* in_sizes, int n_in,
                              void* d_out, int out_size, void* d_ws, size_t ws_size,
                              hipStream_t stream) {
    const float* h      = (const float*)d_in[0];
    const float* dnorm  = (const float*)d_in[1];
    const float* gate_w = (const float*)d_in[2];
    const float* gate_b = (const float*)d_in[3];
    const int*   src    = (const int*)d_in[4];
    const int*   dst    = (const int*)d_in[5];
    float*       z      = (float*)d_out;

    const int n = in_sizes[1];   // N (dnorm element count)
    const int e = in_sizes[4];   // E (src element count)

    float* a = (float*)d_ws;     // N floats
    float* b = a + n;            // N floats  (total 2*N*4 = 800 KB of d_ws)

    // 1) zero output
    {
        int total  = n * DFEAT;
        int blocks = (total + 255) / 256;
        fa_zero_kernel<<<blocks, 256, 0, stream>>>(z, total);
    }

    // 2) node gate projections via WMMA (one wave per 16 nodes)
    {
        int waves   = (n + 15) / 16;
        int threads = 256;                         // 8 waves/block
        int blocks  = (waves * 32 + threads - 1) / threads;
        fa_node_gates_kernel<<<blocks, threads, 0, stream>>>(h, gate_w, a, b, n);
    }

    // 3) per-edge gather + atomic scatter (one wave per edge)
    {
        int waves_per_block = 256 / 32;
        int blocks = (e + waves_per_block - 1) / waves_per_block;
        fa_edge_kernel<<<blocks, 256, 0, stream>>>(h, dnorm, a, b, src, dst,
                                                   gate_b, z, e);
    }
}